// Encoder_53102975648364
// MI455X (gfx1250) — compile-verified
//
#include <hip/hip_runtime.h>

#define HID 256
#define G   (3 * HID)            // 768 gates per matrix
#define TILES_PER_MAT 48         // 768 / 16
#define TOTAL_TILES   96

typedef __attribute__((ext_vector_type(2))) float v2f;
typedef __attribute__((ext_vector_type(8))) float v8f;

// ---------------------------------------------------------------------------
// Kernel 1: 96 single-wave workgroups. Workgroup t computes one 16-row tile of
// gi (t < 48, rows of W_ih against x = embedding[idx]) or gh (t >= 48, rows of
// W_hh against h), using V_WMMA_F32_16X16X4_F32 with B = column-broadcast
// vector chunk, accumulating K=256 in 64 WMMA steps. Bias is fused. Results go
// to gates[] in workspace: [0,768) = gi + b_ih, [768,1536) = gh + b_hh.
// ---------------------------------------------------------------------------
extern "C" __global__ void __launch_bounds__(32)
gru_gemv_wmma(const int*   __restrict__ input,
              const float* __restrict__ hidden,
              const float* __restrict__ embedding,
              const float* __restrict__ w_ih,
              const float* __restrict__ w_hh,
              const float* __restrict__ b_ih,
              const float* __restrict__ b_hh,
              float*       __restrict__ gates)
{
    const int lane = threadIdx.x;        // 0..31 (wave32)
    const int t    = blockIdx.x;         // 0..95 tile id

    const bool   ih = (t < TILES_PER_MAT);
    const float* W  = ih ? w_ih : w_hh;
    const float* v  = ih ? (embedding + (size_t)input[0] * HID) : hidden;
    const float* b  = ih ? b_ih : b_hh;
    const int row_base = (ih ? t : t - TILES_PER_MAT) * 16;

    // A-matrix 16x4 fp32 layout (ISA 7.12.2):
    //   lanes 0-15 : row M=lane,   VGPR0=K0, VGPR1=K1
    //   lanes 16-31: row M=lane-16,VGPR0=K2, VGPR1=K3
    const int m    = lane & 15;          // row within tile
    const int kof  = (lane >> 4) << 1;   // 0 for K={0,1}, 2 for K={2,3}
    const int half = lane >> 4;          // 0 or 1

    const float* wrow = W + (size_t)(row_base + m) * HID + kof;

    v8f acc = {};
    #pragma unroll 8
    for (int k0 = 0; k0 < HID; k0 += 4) {
        v2f a  = *(const v2f*)(wrow + k0);       // W[row][k0+kof .. k0+kof+1]
        v2f bb = *(const v2f*)(v + k0 + kof);    // broadcast vector chunk (B)
        // D = A(16x4) x B(4x16) + C  -- every column of D holds the dot product
        acc = __builtin_amdgcn_wmma_f32_16x16x4_f32(
            false, a, false, bb, (short)0, acc, false, false);
    }

    // C/D layout: VGPR r = row (r + 8*half), identical across lanes of a half.
    // Lane 0 writes rows 0..7, lane 16 writes rows 8..15.
    if (m == 0) {
        float* dst = gates + (ih ? 0 : G);
        #pragma unroll
        for (int r = 0; r < 8; ++r) {
            const int row = row_base + half * 8 + r;
            dst[row] = acc[r] + b[row];
        }
    }
}

// ---------------------------------------------------------------------------
// Kernel 2: fused GRU gate epilogue over 256 elements (1 block, 8 waves).
//   r = sigmoid(i_r + h_r); z = sigmoid(i_z + h_z)
//   n = tanh(i_n + r * h_n); h_new = (1 - z) * n + z * h
// Output tuple (out, out) -> write h_new twice.
// ---------------------------------------------------------------------------
extern "C" __global__ void __launch_bounds__(HID)
gru_gates(const float* __restrict__ gates,
          const float* __restrict__ hidden,
          float*       __restrict__ out)
{
    const int j = threadIdx.x;           // 0..255
    const float ir = gates[j];
    const float iz = gates[HID + j];
    const float in_ = gates[2 * HID + j];
    const float hr = gates[G + j];
    const float hz = gates[G + HID + j];
    const float hn = gates[G + 2 * HID + j];
    const float h  = hidden[j];

    const float r  = 1.0f / (1.0f + __expf(-(ir + hr)));
    const float z  = 1.0f / (1.0f + __expf(-(iz + hz)));
    const float n  = tanhf(in_ + r * hn);
    const float hv = (1.0f - z) * n + z * h;

    out[j]       = hv;
    out[HID + j] = hv;
}

extern "C" void kernel_launch(void* const* d_in, const int* in_sizes, int n_in,
                              void* d_out, int out_size, void* d_ws, size_t ws_size,
                              hipStream_t stream)
{
    (void)in_sizes; (void)n_in; (void)out_size; (void)ws_size;

    const int*   input     = (const int*)  d_in[0];
    const float* hidden    = (const float*)d_in[1];
    const float* embedding = (const float*)d_in[2];
    const float* w_ih      = (const float*)d_in[3];
    const float* w_hh      = (const float*)d_in[4];
    const float* b_ih      = (const float*)d_in[5];
    const float* b_hh      = (const float*)d_in[6];

    float* gates = (float*)d_ws;         // 1536 floats = 6 KB scratch
    float* out   = (float*)d_out;        // 512 floats (h_new duplicated)

    // 96 single-wave workgroups: one 16-row WMMA tile each, spread over WGPs.
    gru_gemv_wmma<<<dim3(TOTAL_TILES), dim3(32), 0, stream>>>(
        input, hidden, embedding, w_ih, w_hh, b_ih, b_hh, gates);

    // Dependent epilogue (stream-ordered): fuse gates, write duplicated out.
    gru_gates<<<dim3(1), dim3(HID), 0, stream>>>(gates, hidden, out);
}